// PumpDetector_14113262535237
// MI455X (gfx1250) — compile-verified
//
#include <hip/hip_runtime.h>

typedef __attribute__((ext_vector_type(2))) float v2f;
typedef __attribute__((ext_vector_type(8))) float v8f;

// ---------------------------------------------------------------------------
// Degree computation: deg starts at 1.0 (self-loop), += 1 per incoming edge,
// then converted in place to deg^{-1/2}.
// ---------------------------------------------------------------------------
__global__ void k_deg_init(float* __restrict__ deg, int n) {
    int i = blockIdx.x * blockDim.x + threadIdx.x;
    if (i < n) deg[i] = 1.0f;  // self loop
}

__global__ void k_deg_accum(const int* __restrict__ dst, float* __restrict__ deg, int E) {
    int e = blockIdx.x * blockDim.x + threadIdx.x;
    if (e < E) atomicAdd(&deg[dst[e]], 1.0f);
}

__global__ void k_dinv(float* __restrict__ deg, int n) {
    int i = blockIdx.x * blockDim.x + threadIdx.x;
    if (i < n) deg[i] = rsqrtf(deg[i]);  // deg >= 1 always
}

// ---------------------------------------------------------------------------
// GEMM1: t1[N,32] = x[N,64] @ W1[64,32] via V_WMMA_F32_16X16X4_F32.
// One wave handles a 16-row tile (full 32-col width -> two 16x16 C tiles).
// A layout (16x4 f32): lane l, g=l>>4, m=m0+(l&15): v0=X[m][k0+2g], v1=X[m][k0+2g+1]
// B layout (4x16 f32): lane l, n=(l&15):            v0=W[k0+2g][n], v1=W[k0+2g+1][n]
// D layout (16x16 f32): D[v] @ lane l = out[m0 + v + 8g][n0 + (l&15)]
// Requires n % 16 == 0 (true: 100000 = 6250*16).
// ---------------------------------------------------------------------------
__global__ void k_gemm1_wmma(const float* __restrict__ x, const float* __restrict__ W,
                             float* __restrict__ t1, int n) {
    int wave = blockIdx.x * 8 + (threadIdx.x >> 5);  // 8 waves / 256-thread block
    int lane = threadIdx.x & 31;
    int m0 = wave * 16;
    if (m0 >= n) return;  // wave-uniform: EXEC all-1s inside
    int g  = lane >> 4;
    int nl = lane & 15;
    const float* xr = x + (size_t)(m0 + nl) * 64;

    v8f c0 = {}, c1 = {};
    #pragma unroll
    for (int k0 = 0; k0 < 64; k0 += 4) {
        int ka = k0 + 2 * g;
        v2f a;  a.x  = xr[ka];              a.y  = xr[ka + 1];
        v2f b0; b0.x = W[ka * 32 + nl];     b0.y = W[(ka + 1) * 32 + nl];
        v2f b1; b1.x = W[ka * 32 + 16 + nl]; b1.y = W[(ka + 1) * 32 + 16 + nl];
        c0 = __builtin_amdgcn_wmma_f32_16x16x4_f32(false, a, false, b0, (short)0, c0, false, false);
        c1 = __builtin_amdgcn_wmma_f32_16x16x4_f32(false, a, false, b1, (short)0, c1, false, false);
    }
    #pragma unroll
    for (int v = 0; v < 8; ++v) {
        int row = m0 + v + 8 * g;
        t1[(size_t)row * 32 + nl]      = c0[v];
        t1[(size_t)row * 32 + 16 + nl] = c1[v];
    }
}

// ---------------------------------------------------------------------------
// GEMM2: t2[N,16] = relu(acc1)[N,32] @ W2[32,16]; relu fused into A load.
// ---------------------------------------------------------------------------
__global__ void k_gemm2_wmma(const float* __restrict__ acc1, const float* __restrict__ W,
                             float* __restrict__ t2, int n) {
    int wave = blockIdx.x * 8 + (threadIdx.x >> 5);
    int lane = threadIdx.x & 31;
    int m0 = wave * 16;
    if (m0 >= n) return;
    int g  = lane >> 4;
    int nl = lane & 15;
    const float* hr = acc1 + (size_t)(m0 + nl) * 32;

    v8f c = {};
    #pragma unroll
    for (int k0 = 0; k0 < 32; k0 += 4) {
        int ka = k0 + 2 * g;
        v2f a; a.x = fmaxf(hr[ka], 0.0f);  a.y = fmaxf(hr[ka + 1], 0.0f);
        v2f b; b.x = W[ka * 16 + nl];      b.y = W[(ka + 1) * 16 + nl];
        c = __builtin_amdgcn_wmma_f32_16x16x4_f32(false, a, false, b, (short)0, c, false, false);
    }
    #pragma unroll
    for (int v = 0; v < 8; ++v)
        t2[(size_t)(m0 + v + 8 * g) * 16 + nl] = c[v];
}

// ---------------------------------------------------------------------------
// acc[i,c] = bias[c] + t[i,c] * dinv[i]^2   (bias + self-loop contribution)
// d = 1<<shift channels per node.
// ---------------------------------------------------------------------------
__global__ void k_init_acc(const float* __restrict__ t, const float* __restrict__ dinv,
                           const float* __restrict__ bias, float* __restrict__ acc,
                           int n, int shift) {
    int tid = blockIdx.x * blockDim.x + threadIdx.x;
    int i = tid >> shift;
    int c = tid & ((1 << shift) - 1);
    if (i < n) {
        float w = dinv[i];
        acc[tid] = bias[c] + t[tid] * w * w;
    }
}

// ---------------------------------------------------------------------------
// Edge scatter: one lane per (edge, channel); a wave covers all channels of
// one (d=32) or two (d=16) edges -> coalesced gather + f32 atomic scatter.
// ---------------------------------------------------------------------------
__global__ void k_edge_scatter(const int* __restrict__ src, const int* __restrict__ dst,
                               const float* __restrict__ t, const float* __restrict__ dinv,
                               float* __restrict__ acc, int E, int shift) {
    long long tid = (long long)blockIdx.x * blockDim.x + threadIdx.x;
    long long e = tid >> shift;
    int c = (int)(tid & ((1 << shift) - 1));
    if (e < E) {
        int s = src[e];
        int d = dst[e];
        float norm = dinv[s] * dinv[d];
        float v = t[((size_t)s << shift) + c] * norm;
        atomicAdd(&acc[((size_t)d << shift) + c], v);
    }
}

// ---------------------------------------------------------------------------
// out[i] = sigmoid( relu(acc2[i,:]) . Wc + bc )
// ---------------------------------------------------------------------------
__global__ void k_final(const float* __restrict__ acc2, const float* __restrict__ Wc,
                        const float* __restrict__ bc, float* __restrict__ out, int n) {
    int i = blockIdx.x * blockDim.x + threadIdx.x;
    if (i < n) {
        float z = bc[0];
        const float* h = acc2 + (size_t)i * 16;
        #pragma unroll
        for (int c = 0; c < 16; ++c)
            z += fmaxf(h[c], 0.0f) * Wc[c];
        out[i] = 1.0f / (1.0f + __expf(-z));
    }
}

// ---------------------------------------------------------------------------
extern "C" void kernel_launch(void* const* d_in, const int* in_sizes, int n_in,
                              void* d_out, int out_size, void* d_ws, size_t ws_size,
                              hipStream_t stream) {
    const float* x  = (const float*)d_in[0];
    const int*   ei = (const int*)d_in[1];
    const float* W1 = (const float*)d_in[2];
    const float* b1 = (const float*)d_in[3];
    const float* W2 = (const float*)d_in[4];
    const float* b2 = (const float*)d_in[5];
    const float* Wc = (const float*)d_in[6];
    const float* bc = (const float*)d_in[7];
    float* out = (float*)d_out;

    const int n = in_sizes[0] / 64;   // 100000
    const int E = in_sizes[1] / 2;    // 1250000
    const int* src = ei;              // edge_index[0]
    const int* dst = ei + E;          // edge_index[1]

    // Workspace layout (floats): dinv[n] | t1[32n] | acc1[32n] | t2[16n] | acc2[16n]
    float* ws   = (float*)d_ws;
    float* dinv = ws;
    float* t1   = dinv + n;
    float* acc1 = t1 + (size_t)32 * n;
    float* t2   = acc1 + (size_t)32 * n;
    float* acc2 = t2 + (size_t)16 * n;

    const int TB = 256;
    // --- degrees ---
    k_deg_init <<<(n + TB - 1) / TB, TB, 0, stream>>>(dinv, n);
    k_deg_accum<<<(E + TB - 1) / TB, TB, 0, stream>>>(dst, dinv, E);
    k_dinv     <<<(n + TB - 1) / TB, TB, 0, stream>>>(dinv, n);

    // --- layer 1 ---
    int waves  = (n + 15) / 16;            // 16-row tiles, one wave each
    int gblks  = (waves + 7) / 8;          // 8 waves per 256-thread block
    k_gemm1_wmma<<<gblks, TB, 0, stream>>>(x, W1, t1, n);
    k_init_acc  <<<((size_t)32 * n + TB - 1) / TB, TB, 0, stream>>>(t1, dinv, b1, acc1, n, 5);
    long long tot1 = (long long)E << 5;
    k_edge_scatter<<<(unsigned)((tot1 + TB - 1) / TB), TB, 0, stream>>>(src, dst, t1, dinv, acc1, E, 5);

    // --- layer 2 ---
    k_gemm2_wmma<<<gblks, TB, 0, stream>>>(acc1, W2, t2, n);
    k_init_acc  <<<((size_t)16 * n + TB - 1) / TB, TB, 0, stream>>>(t2, dinv, b2, acc2, n, 4);
    long long tot2 = (long long)E << 4;
    k_edge_scatter<<<(unsigned)((tot2 + TB - 1) / TB), TB, 0, stream>>>(src, dst, t2, dinv, acc2, E, 4);

    // --- classifier ---
    k_final<<<(n + TB - 1) / TB, TB, 0, stream>>>(acc2, Wc, bc, out, n);
}